// GMDTW_57475252355413
// MI455X (gfx1250) — compile-verified
//
#include <hip/hip_runtime.h>
#include <hip/hip_bf16.h>
#include <math.h>

// ---------------------------------------------------------------------------
// Soft-DTW (GMDTW v2), m1 = m2 = 2048, d = 128, gamma = 1.0
//
// Phase 1: norms of rows of y / x2                     (norms_kernel)
// Phase 2: C = n1[i] + n2[j] - 2*y@x2^T via fp32 WMMA  (gemm_kernel)
//          stored DIAGONAL-MAJOR: Cd[(i+j)*2048 + i]
// Phase 3: serial anti-diagonal DP in LDS, 1 workgroup (dp_kernel)
// ---------------------------------------------------------------------------

#define M1   2048
#define M2   2048
#define DIM  128

typedef float v2f __attribute__((ext_vector_type(2)));
typedef float v8f __attribute__((ext_vector_type(8)));

// ---------------------------------------------------------------- norms ----
__global__ void __launch_bounds__(256)
norms_kernel(const float* __restrict__ x1, const float* __restrict__ x2,
             float* __restrict__ n1, float* __restrict__ n2) {
    int r = blockIdx.x * blockDim.x + threadIdx.x;      // 0..4095
    const float* src = (r < M1) ? (x1 + (size_t)r * DIM)
                                : (x2 + (size_t)(r - M1) * DIM);
    float s = 0.0f;
#pragma unroll
    for (int k = 0; k < DIM; k += 4) {
        float4 v = *(const float4*)(src + k);
        s += v.x * v.x + v.y * v.y + v.z * v.z + v.w * v.w;
    }
    if (r < M1) n1[r] = s; else n2[r - M1] = s;
}

// ----------------------------------------------------------------- gemm ----
// One wave per 16x16 output tile. 128x128 tiles -> 16384 waves -> 2048 blocks.
__global__ void __launch_bounds__(256)
gemm_kernel(const float* __restrict__ x1, const float* __restrict__ x2,
            const float* __restrict__ n1, const float* __restrict__ n2,
            float* __restrict__ Cd) {
    int wave = (blockIdx.x * 256 + threadIdx.x) >> 5;
    int lane = threadIdx.x & 31;
    int ti = wave >> 7;            // tile row  0..127
    int tj = wave & 127;           // tile col  0..127
    int i0 = ti << 4, j0 = tj << 4;
    int m    = lane & 15;          // row (A) / col (B) within tile
    int half = lane >> 4;          // K sub-pair selector

    const float* arow = x1 + (size_t)(i0 + m) * DIM + 2 * half;
    const float* brow = x2 + (size_t)(j0 + m) * DIM + 2 * half;

    v8f acc = {};
#pragma unroll
    for (int kk = 0; kk < DIM / 4; ++kk) {
        // A 16x4 f32: lane m: VGPR0=K(2*half), VGPR1=K(2*half+1)   (contig float2)
        v2f a = *(const v2f*)(arow + kk * 4);
        // B 4x16 f32 (= x2^T tile): same per-lane pattern from x2 rows
        v2f b = *(const v2f*)(brow + kk * 4);
        acc = __builtin_amdgcn_wmma_f32_16x16x4_f32(
            /*neg_a=*/false, a, /*neg_b=*/false, b,
            /*c_mod=*/(short)0, acc, /*reuse_a=*/false, /*reuse_b=*/false);
    }

    // C/D layout: VGPR r, lane L -> row r + 8*(L>>4), col L&15
    float norm2v = n2[j0 + m];
#pragma unroll
    for (int r = 0; r < 8; ++r) {
        int gi = i0 + r + 8 * half;
        int gj = j0 + m;
        float c = n1[gi] + norm2v - 2.0f * acc[r];
        // diagonal-major: diag kk = gi+gj (0..4094), index gi within diag
        Cd[(size_t)(gi + gj) * M1 + gi] = c;
    }
}

// ------------------------------------------------------------------- dp ----
// R[i,j] = C[i-1,j-1] + softmin_1(R[i-1,j], R[i,j-1], R[i-1,j-1])
// Single workgroup (32 waves), rotating diagonals in LDS, barrier per diag.
__global__ void __launch_bounds__(1024)
dp_kernel(const float* __restrict__ Cd, float* __restrict__ out) {
    __shared__ float buf[3][M1 + 1];
    const int tid = threadIdx.x;

    for (int i = tid; i <= M1; i += 1024) {
        buf[0][i] = (i == 0) ? 0.0f : INFINITY;   // diag k=0
        buf[1][i] = INFINITY;                     // diag k=1
        buf[2][i] = INFINITY;
    }
    __syncthreads();

    for (int k = 2; k <= M1 + M2; ++k) {
        const float* rp2 = buf[(k - 2) % 3];
        const float* rp1 = buf[(k - 1) % 3];
        float*       rn  = buf[k % 3];

        // prefetch next diagonal's C row (coalesced, L2-resident)
        if (k <= M1 + M2 - 1) {
            int pi = tid + 1;
            int pj = (k + 1) - pi;
            if (pj >= 1 && pj <= M2)
                __builtin_prefetch(&Cd[(size_t)(k - 1) * M1 + (pi - 1)], 0, 0);
        }

#pragma unroll
        for (int c = 0; c < 2; ++c) {
            int i = tid + 1 + c * 1024;           // 1..2048
            int j = k - i;
            float r = INFINITY;
            if (j >= 1 && j <= M2) {
                float a = rp1[i - 1];             // R[i-1, j]
                float b = rp1[i];                 // R[i,   j-1]
                float d = rp2[i - 1];             // R[i-1, j-1]
                float mn = fminf(fminf(a, b), d); // finite for all valid cells
                float s  = __expf(mn - a) + __expf(mn - b) + __expf(mn - d);
                float softmin = mn - __logf(s);
                float cv = Cd[(size_t)(k - 2) * M1 + (i - 1)];
                r = cv + softmin;
                if (k == M1 + M2 && i == M1) out[0] = r;   // R[m1, m2]
            }
            rn[i] = r;                            // borders/invalid -> +inf
        }
        if (tid == 0) rn[0] = INFINITY;
        __syncthreads();
    }
}

// ------------------------------------------------------------- launcher ----
extern "C" void kernel_launch(void* const* d_in, const int* in_sizes, int n_in,
                              void* d_out, int out_size, void* d_ws, size_t ws_size,
                              hipStream_t stream) {
    const float* y  = (const float*)d_in[0];   // (2048, 128) fp32
    const float* x2 = (const float*)d_in[1];   // (2048, 128) fp32

    float* ws = (float*)d_ws;
    float* n1 = ws;                 // 2048 floats
    float* n2 = ws + M1;            // 2048 floats
    float* Cd = ws + 2 * M1;        // 4095 * 2048 floats (diag-major cost)

    norms_kernel<<<(M1 + M2) / 256, 256, 0, stream>>>(y, x2, n1, n2);
    gemm_kernel <<<(M1 / 16) * (M2 / 16) / 8, 256, 0, stream>>>(y, x2, n1, n2, Cd);
    dp_kernel   <<<1, 1024, 0, stream>>>(Cd, (float*)d_out);
}